// UnifiedQuantumRegressor_65481071402742
// MI455X (gfx1250) — compile-verified
//
#include <hip/hip_runtime.h>
#include <hip/hip_bf16.h>
#include <math.h>

// Problem constants (from the reference)
#define NROWS 8192
#define FIN   512
#define HH    256
#define WW    64

typedef __attribute__((ext_vector_type(2))) float v2f;
typedef __attribute__((ext_vector_type(8))) float v8f;

// One wave computes a 16x16 fp32 tile D += A(16xK) * B(Kx16) using
// V_WMMA_F32_16X16X4_F32, looping K in steps of 4.
//
// Per the CDNA5 ISA VGPR layouts (CDNA5 ISA):
//   A 16x4  : lanes 0-15 -> M=lane, v0=K0 v1=K1 ; lanes 16-31 -> v0=K2 v1=K3
//   B 4x16  : v0 = rows K0 (lanes 0-15) / K2 (lanes 16-31), v1 = K1 / K3
//   C/D     : VGPR i -> (M = i + 8*half, N = lane&15)
__device__ __forceinline__ v8f gemm16_f32(const float* __restrict__ Abase, int lda,
                                          const float* __restrict__ Bbase, int ldb,
                                          int K, v8f c) {
    const int lane = threadIdx.x & 31;
    const int l    = lane & 15;
    const int half = lane >> 4;
    const float* arow = Abase + l * lda + 2 * half;  // row (l) of the tile, half-split K
    #pragma unroll 4
    for (int kk = 0; kk < K; kk += 4) {
        v2f a;
        a[0] = arow[kk];
        a[1] = arow[kk + 1];
        v2f b;
        b[0] = Bbase[(kk + 2 * half)     * ldb + l];
        b[1] = Bbase[(kk + 2 * half + 1) * ldb + l];
        c = __builtin_amdgcn_wmma_f32_16x16x4_f32(
                /*neg_a=*/false, a, /*neg_b=*/false, b,
                /*c_mod=*/(short)0, c, /*reuse_a=*/false, /*reuse_b=*/false);
    }
    return c;
}

// Fused 3-layer MLP over a 16-row slab of the batch.
// grid.x = NROWS/16 blocks, 512 threads = 16 waves per block.
//   phase 1: wave w computes enc1 tile (cols 16w..16w+15), K=512, relu  -> LDS
//   phase 2: wave w computes enc2 tile,                    K=256, relu  -> LDS
//   phase 3: waves 0..3 compute q tile (W=64),             K=256, tanh,
//            reduce column sums within the tile and emit per-block partials.
__global__ __launch_bounds__(512)
void fused_mlp_kernel(const float* __restrict__ state,
                      const float* __restrict__ W1, const float* __restrict__ b1,
                      const float* __restrict__ W2, const float* __restrict__ b2,
                      const float* __restrict__ Wq, const float* __restrict__ bq,
                      float* __restrict__ partial /* [gridDim.x * WW] */) {
    __shared__ float enc1[16 * HH];
    __shared__ float enc2[16 * HH];

    const int wave = threadIdx.x >> 5;
    const int lane = threadIdx.x & 31;
    const int l    = lane & 15;
    const int half = lane >> 4;
    const int m0   = blockIdx.x * 16;

    // Warm L2/WGP$ for the next layers' weights while phase 1 runs.
    __builtin_prefetch(W2 + threadIdx.x * 16, 0, 0);
    __builtin_prefetch(Wq + threadIdx.x * 8, 0, 0);

    // ---- phase 1: enc1 = relu(state @ W1 + b1) --------------------------
    {
        const int n0 = wave * 16;
        v8f c = {};
        c = gemm16_f32(state + (size_t)m0 * FIN, FIN, W1 + n0, HH, FIN, c);
        const float bias = b1[n0 + l];
        #pragma unroll
        for (int i = 0; i < 8; ++i) {
            float v = c[i] + bias;
            enc1[(i + 8 * half) * HH + n0 + l] = v > 0.f ? v : 0.f;
        }
    }
    __syncthreads();

    // ---- phase 2: enc2 = relu(enc1 @ W2 + b2) ---------------------------
    {
        const int n0 = wave * 16;
        v8f c = {};
        c = gemm16_f32(enc1, HH, W2 + n0, HH, HH, c);
        const float bias = b2[n0 + l];
        #pragma unroll
        for (int i = 0; i < 8; ++i) {
            float v = c[i] + bias;
            enc2[(i + 8 * half) * HH + n0 + l] = v > 0.f ? v : 0.f;
        }
    }
    __syncthreads();

    // ---- phase 3: q = tanh(enc2 @ Wq + bq); per-block column sums -------
    if (wave < 4) {                      // wave-uniform branch: EXEC all-1s
        const int n0 = wave * 16;
        v8f c = {};
        c = gemm16_f32(enc2, HH, Wq + n0, WW, HH, c);
        const float bias = bq[n0 + l];
        float colsum = 0.f;
        #pragma unroll
        for (int i = 0; i < 8; ++i)
            colsum += tanhf(c[i] + bias);
        // add the other half-tile's rows (lanes l and l+16 hold same column)
        colsum += __shfl_xor(colsum, 16, 32);
        if (half == 0)
            partial[blockIdx.x * WW + n0 + l] = colsum;
    }
}

// Deterministic finalize: s[w] = sum_b partial[b][w]; out = dot(s, Wh) + bh
__global__ __launch_bounds__(64)
void finalize_kernel(const float* __restrict__ partial, int nblocks,
                     const float* __restrict__ Wh, const float* __restrict__ bh,
                     float* __restrict__ out) {
    __shared__ float red[64];
    const int w = threadIdx.x;           // 0..63
    float s = 0.f;
    for (int b = 0; b < nblocks; ++b)
        s += partial[b * WW + w];
    red[w] = s * Wh[w];
    __syncthreads();
    #pragma unroll
    for (int off = 32; off > 0; off >>= 1) {
        if (w < off) red[w] += red[w + off];
        __syncthreads();
    }
    if (w == 0) out[0] = red[0] + bh[0];
}

extern "C" void kernel_launch(void* const* d_in, const int* in_sizes, int n_in,
                              void* d_out, int out_size, void* d_ws, size_t ws_size,
                              hipStream_t stream) {
    const float* state = (const float*)d_in[0];
    const float* W1    = (const float*)d_in[1];
    const float* b1    = (const float*)d_in[2];
    const float* W2    = (const float*)d_in[3];
    const float* b2    = (const float*)d_in[4];
    const float* Wq    = (const float*)d_in[5];
    const float* bq    = (const float*)d_in[6];
    const float* Wh    = (const float*)d_in[7];
    const float* bh    = (const float*)d_in[8];
    float* out     = (float*)d_out;
    float* partial = (float*)d_ws;       // 512 * 64 * 4B = 128 KB scratch

    const int nblocks = NROWS / 16;      // 512
    fused_mlp_kernel<<<nblocks, 512, 0, stream>>>(state, W1, b1, W2, b2, Wq, bq, partial);
    finalize_kernel<<<1, 64, 0, stream>>>(partial, nblocks, Wh, bh, out);
}